// Generator3DLUT_zero_43130061586455
// MI455X (gfx1250) — compile-verified
//
#include <hip/hip_runtime.h>

// 3D-LUT trilinear interpolation, tuned for MI455X (gfx1250):
//  - memory-bound: 192 MiB streamed at 23.3 TB/s (~8.6 us floor)
//  - 431 KB LUT kept resident in the 192 MB L2 (regular-temporal gathers)
//  - frame planes streamed with non-temporal B128 loads/stores + global_prefetch_b8
//  - wave32-friendly 256-thread blocks (8 waves)

typedef float v4f __attribute__((ext_vector_type(4)));

constexpr int   LUT_DIM  = 33;
constexpr int   LUT_DIM2 = LUT_DIM * LUT_DIM;      // 1089
constexpr int   LUT_DIM3 = LUT_DIM * LUT_DIM2;     // 35937
constexpr int   NPIX     = 1024 * 1024;            // pixels per plane
constexpr int   NQ       = NPIX / 4;               // float4 quads per plane (2^18)
constexpr float INV_BIN  = 32.0f / 1.000001f;      // 1 / binsize

__device__ __forceinline__ void lut_px(const float* __restrict__ lut,
                                       float r, float g, float b,
                                       float& oR, float& oG, float& oB)
{
    float rs = r * INV_BIN, gs = g * INV_BIN, bs = b * INV_BIN;
    int rid = (int)floorf(rs); rid = min(max(rid, 0), LUT_DIM - 2);
    int gid = (int)floorf(gs); gid = min(max(gid, 0), LUT_DIM - 2);
    int bid = (int)floorf(bs); bid = min(max(bid, 0), LUT_DIM - 2);
    float rd = rs - (float)rid;
    float gd = gs - (float)gid;
    float bd = bs - (float)bid;
    float wr0   = 1.0f - rd;
    float wg[2] = {1.0f - gd, gd};
    float wb[2] = {1.0f - bd, bd};

    int base = rid + gid * LUT_DIM + bid * LUT_DIM2;
    float aR = 0.0f, aG = 0.0f, aB = 0.0f;
#pragma unroll
    for (int db = 0; db < 2; ++db) {
#pragma unroll
        for (int dg = 0; dg < 2; ++dg) {
            const float* __restrict__ p = lut + (base + dg * LUT_DIM + db * LUT_DIM2);
            float w = wg[dg] * wb[db];
            // dr=0 / dr=1 corners are adjacent in memory for each channel
            float c00 = p[0],            c01 = p[1];
            float c10 = p[LUT_DIM3],     c11 = p[LUT_DIM3 + 1];
            float c20 = p[2 * LUT_DIM3], c21 = p[2 * LUT_DIM3 + 1];
            aR = fmaf(w, fmaf(rd, c01, wr0 * c00), aR);
            aG = fmaf(w, fmaf(rd, c11, wr0 * c10), aG);
            aB = fmaf(w, fmaf(rd, c21, wr0 * c20), aB);
        }
    }
    oR = aR; oG = aG; oB = aB;
}

__global__ __launch_bounds__(256) void lut3d_apply(const float* __restrict__ lut,
                                                   const float* __restrict__ x,
                                                   float* __restrict__ out,
                                                   int nq_total)
{
    int i4 = blockIdx.x * blockDim.x + threadIdx.x;
    if (i4 >= nq_total) return;
    int img = i4 >> 18;          // i4 / NQ   (NQ == 2^18)
    int q   = i4 & (NQ - 1);     // quad index within one plane

    const v4f* __restrict__ xin = (const v4f*)(x   + (size_t)img * 3 * NPIX);
    v4f*       __restrict__ op  = (v4f*)      (out + (size_t)img * 3 * NPIX);

    // Speculative prefetch ~64 KB ahead on each streaming plane.
    // gfx1250: lowers to global_prefetch_b8; translation faults are silently dropped.
    __builtin_prefetch(&xin[q + 4096], 0, 1);
    __builtin_prefetch(&xin[q + NQ + 4096], 0, 1);
    __builtin_prefetch(&xin[q + 2 * NQ + 4096], 0, 1);

    // Read-once streams: non-temporal so 192 MiB of frame traffic doesn't evict
    // the L2-resident LUT / pollute the cache hierarchy.
    v4f r4 = __builtin_nontemporal_load(&xin[q]);
    v4f g4 = __builtin_nontemporal_load(&xin[q + NQ]);
    v4f b4 = __builtin_nontemporal_load(&xin[q + 2 * NQ]);

    v4f oR, oG, oB;
#pragma unroll
    for (int k = 0; k < 4; ++k) {
        float tR, tG, tB;
        lut_px(lut, r4[k], g4[k], b4[k], tR, tG, tB);
        oR[k] = tR;
        oG[k] = tG;
        oB[k] = tB;
    }

    __builtin_nontemporal_store(oR, &op[q]);
    __builtin_nontemporal_store(oG, &op[q + NQ]);
    __builtin_nontemporal_store(oB, &op[q + 2 * NQ]);
}

extern "C" void kernel_launch(void* const* d_in, const int* in_sizes, int n_in,
                              void* d_out, int out_size, void* d_ws, size_t ws_size,
                              hipStream_t stream)
{
    const float* lut = (const float*)d_in[0];   // (3, 33, 33, 33) fp32
    const float* x   = (const float*)d_in[1];   // (8, 3, 1024, 1024) fp32
    float*       out = (float*)d_out;           // (8, 3, 1024, 1024) fp32

    // total float4 quads across all images/planes handled per thread:
    // in_sizes[1] = B*3*NPIX ; each thread does one quad of all 3 channels
    int nq_total = in_sizes[1] / 12;
    int blocks   = (nq_total + 255) / 256;
    lut3d_apply<<<blocks, 256, 0, stream>>>(lut, x, out, nq_total);
}